// MUSIC_39805756899659
// MI455X (gfx1250) — compile-verified
//
#include <hip/hip_runtime.h>
#include <math.h>

typedef float v2f __attribute__((ext_vector_type(2)));
typedef float v8f __attribute__((ext_vector_type(8)));

#define N_ANGLES 18001
#define N_FREQ 19      // rfft bins 13..31
#define K0_BIN 13
#define EPS_MUSIC 1e-8f

// ---------------------------------------------------------------------------
// Kernel 1: DFT of the 16 (mic,frame) windowed frames at bins 13..31 via
// V_WMMA_F32_16X16X4_F32.  Output tile columns j: bin k = 13 + j/2,
// comp = j&1 (0 = sum*cos = Re X, 1 = sum*sin = -Im X).
// Twiddles come from an LDS table indexed by (k*n) mod 512 (exact reduction),
// advanced incrementally -> inner loop is pure LDS + WMMA.
// ---------------------------------------------------------------------------
__global__ __launch_bounds__(128) void music_dft_wmma(
    const float* __restrict__ x, float* __restrict__ ws_X) {
  __shared__ float fr[16 * 512];          // 32 KB: windowed frames, m = c*4+t
  __shared__ float ct[512], st[512];      // cos/sin(pi*i/256)

  const int tid = threadIdx.x;
  const float PI_F = 3.14159265358979323846f;
  const float PIO256 = PI_F / 256.0f;

  for (int i = tid; i < 512; i += 128) {
    float s, c;
    sincosf(PIO256 * (float)i, &s, &c);
    ct[i] = c;
    st[i] = s;
  }
  // Stage: frame t_global = 120+t uses samples (119+t)*256 + n, channel c.
  for (int i = tid; i < 16 * 512; i += 128) {
    int m = i >> 9;            // 0..15
    int n = i & 511;
    int c = m >> 2;
    int t = m & 3;
    long s = (long)(119 + t) * 256 + n;
    float w = sinf(PI_F * (float)n * (1.0f / 512.0f));  // sqrt(hann) == sin
    fr[i] = x[s * 4 + c] * w;
  }
  __syncthreads();

  const int wave = tid >> 5;
  const int lane = tid & 31;
  if (wave < 3) {
    const int m    = lane & 15;   // A row (M)
    const int half = lane >> 4;   // lane-half selects K pair {0,1} vs {2,3}
    const int jloc = lane & 15;   // B/D column (N)
    int j = wave * 16 + jloc;     // global column
    int kb = K0_BIN + (j >> 1);
    if (kb > 31) kb = 31;         // padded columns (j>=38), never read back
    const float* tbl = (j & 1) ? st : ct;

    const float* frm = &fr[m * 512 + 2 * half];  // A source row
    int idx = (kb * 2 * half) & 511;             // (kb*nb) mod 512 at kk=0
    const int stepi = (4 * kb) & 511;            // advance per K-iteration

    v8f acc = {};
    for (int kk = 0; kk < 512; kk += 4) {
      v2f a;
      a.x = frm[kk];
      a.y = frm[kk + 1];
      v2f b;
      b.x = tbl[idx];
      b.y = tbl[(idx + kb) & 511];
      idx = (idx + stepi) & 511;
      acc = __builtin_amdgcn_wmma_f32_16x16x4_f32(
          /*neg_a=*/false, a, /*neg_b=*/false, b,
          /*c_mod=*/(short)0, acc, /*reuse_a=*/false, /*reuse_b=*/false);
    }
    if (j < 2 * N_FREQ) {
      for (int v = 0; v < 8; ++v) {
        int row = v + 8 * half;            // D row M
        ws_X[j * 16 + row] = acc[v];
      }
    }
  }
}

// ---------------------------------------------------------------------------
// Kernel 2: per frequency — 4x4 Hermitian covariance, complex Jacobi eigh,
// noise projector G = En En^H over the 2 smallest eigenvalues, then emit
// {tr(G), h1, h2, h3} (h_d = sum of d-th subdiagonal of G).
// ---------------------------------------------------------------------------
__global__ __launch_bounds__(32) void music_eig(
    const float* __restrict__ ws_X, float* __restrict__ ws_h) {
  int f = threadIdx.x;
  if (f >= N_FREQ) return;

  double Xr[4][4], Xi[4][4];  // [c][t]
  for (int c = 0; c < 4; ++c)
    for (int t = 0; t < 4; ++t) {
      int m = c * 4 + t;
      Xr[c][t] = (double)ws_X[(2 * f) * 16 + m];
      Xi[c][t] = -(double)ws_X[(2 * f + 1) * 16 + m];  // X = C - i*S
    }

  // R = (1/4) sum_t X[:,t] X[:,t]^H
  double ar[4][4], ai[4][4];
  for (int c = 0; c < 4; ++c)
    for (int d = 0; d < 4; ++d) {
      double rr = 0.0, ii = 0.0;
      for (int t = 0; t < 4; ++t) {
        rr += Xr[c][t] * Xr[d][t] + Xi[c][t] * Xi[d][t];
        ii += Xi[c][t] * Xr[d][t] - Xr[c][t] * Xi[d][t];
      }
      ar[c][d] = 0.25 * rr;
      ai[c][d] = 0.25 * ii;
    }

  double vr[4][4] = {}, vi[4][4] = {};
  vr[0][0] = vr[1][1] = vr[2][2] = vr[3][3] = 1.0;

  for (int sweep = 0; sweep < 8; ++sweep) {
    for (int p = 0; p < 3; ++p)
      for (int q = p + 1; q < 4; ++q) {
        double rpq = ar[p][q], ipq = ai[p][q];
        double r = sqrt(rpq * rpq + ipq * ipq);
        if (r < 1e-300) continue;
        double er = rpq / r, ei = ipq / r;  // e^{i alpha}
        double tau = (ar[q][q] - ar[p][p]) / (2.0 * r);
        double tt = (tau >= 0.0 ? 1.0 : -1.0) / (fabs(tau) + sqrt(1.0 + tau * tau));
        double cc = 1.0 / sqrt(1.0 + tt * tt);
        double ss = tt * cc;
        // A <- A J  (columns p,q);  V <- V J
        for (int k = 0; k < 4; ++k) {
          double apr = ar[k][p], api = ai[k][p];
          double aqr = ar[k][q], aqi = ai[k][q];
          double tqr = er * aqr + ei * aqi, tqi = er * aqi - ei * aqr;  // e^{-ia}*a_kq
          double tpr = er * apr - ei * api, tpi = er * api + ei * apr;  // e^{+ia}*a_kp
          ar[k][p] = cc * apr - ss * tqr;  ai[k][p] = cc * api - ss * tqi;
          ar[k][q] = ss * tpr + cc * aqr;  ai[k][q] = ss * tpi + cc * aqi;

          double wpr = vr[k][p], wpi = vi[k][p];
          double wqr = vr[k][q], wqi = vi[k][q];
          double uqr = er * wqr + ei * wqi, uqi = er * wqi - ei * wqr;
          double upr = er * wpr - ei * wpi, upi = er * wpi + ei * wpr;
          vr[k][p] = cc * wpr - ss * uqr;  vi[k][p] = cc * wpi - ss * uqi;
          vr[k][q] = ss * upr + cc * wqr;  vi[k][q] = ss * upi + cc * wqi;
        }
        // A <- J^H A  (rows p,q)
        for (int k = 0; k < 4; ++k) {
          double apr = ar[p][k], api = ai[p][k];
          double aqr = ar[q][k], aqi = ai[q][k];
          double tqr = er * aqr - ei * aqi, tqi = er * aqi + ei * aqr;  // e^{+ia}*a_qk
          double tpr = er * apr + ei * api, tpi = er * api - ei * apr;  // e^{-ia}*a_pk
          ar[p][k] = cc * apr - ss * tqr;  ai[p][k] = cc * api - ss * tqi;
          ar[q][k] = ss * tpr + cc * aqr;  ai[q][k] = ss * tpi + cc * aqi;
        }
      }
  }

  // two smallest eigenvalues -> noise subspace
  double ev[4] = {ar[0][0], ar[1][1], ar[2][2], ar[3][3]};
  int idx[4] = {0, 1, 2, 3};
  for (int a = 0; a < 3; ++a)
    for (int b = a + 1; b < 4; ++b)
      if (ev[idx[b]] < ev[idx[a]]) { int t_ = idx[a]; idx[a] = idx[b]; idx[b] = t_; }

  double gr[4][4] = {}, gi[4][4] = {};
  for (int s = 0; s < 2; ++s) {
    int col = idx[s];
    for (int c = 0; c < 4; ++c)
      for (int d = 0; d < 4; ++d) {
        gr[c][d] += vr[c][col] * vr[d][col] + vi[c][col] * vi[d][col];
        gi[c][d] += vi[c][col] * vr[d][col] - vr[c][col] * vi[d][col];
      }
  }

  ws_h[f * 8 + 0] = (float)(gr[0][0] + gr[1][1] + gr[2][2] + gr[3][3]);
  ws_h[f * 8 + 1] = 0.0f;
  ws_h[f * 8 + 2] = (float)(gr[1][0] + gr[2][1] + gr[3][2]);
  ws_h[f * 8 + 3] = (float)(gi[1][0] + gi[2][1] + gi[3][2]);
  ws_h[f * 8 + 4] = (float)(gr[2][0] + gr[3][1]);
  ws_h[f * 8 + 5] = (float)(gi[2][0] + gi[3][1]);
  ws_h[f * 8 + 6] = (float)(gr[3][0]);
  ws_h[f * 8 + 7] = (float)(gi[3][0]);
}

// ---------------------------------------------------------------------------
// Kernel 3: 18001-angle spectrum scan.
// denom(phi) = tr + 2*sum_d [Re h_d cos(d phi) + Im h_d sin(d phi)] + eps
// ---------------------------------------------------------------------------
__global__ __launch_bounds__(256) void music_scan(
    const float* __restrict__ ws_h, float* __restrict__ out) {
  __shared__ float h[N_FREQ * 8];
  for (int i = threadIdx.x; i < N_FREQ * 8; i += blockDim.x) h[i] = ws_h[i];
  __syncthreads();

  int a = blockIdx.x * blockDim.x + threadIdx.x;
  if (a >= N_ANGLES) return;

  double theta = ((double)a * 0.01 - 90.0) * (3.14159265358979323846 / 180.0);
  float st = (float)sin(theta);
  // 2*pi * d/c * 31.25 * sin(theta); phi_f = base*(13+f)
  float base = (float)(2.0 * 3.14159265358979323846 * 0.1 / 343.0 * 31.25) * st;

  float acc = 0.0f;
  for (int f = 0; f < N_FREQ; ++f) {
    float phi = base * (float)(K0_BIN + f);
    float s1, c1;
    sincosf(phi, &s1, &c1);
    float c2 = c1 * c1 - s1 * s1, s2 = 2.0f * s1 * c1;
    float c3 = c1 * c2 - s1 * s2, s3 = s1 * c2 + c1 * s2;
    const float* hf = &h[f * 8];
    float denom = hf[0] +
                  2.0f * (hf[2] * c1 + hf[3] * s1 +
                          hf[4] * c2 + hf[5] * s2 +
                          hf[6] * c3 + hf[7] * s3) +
                  EPS_MUSIC;
    acc += 1.0f / denom;
  }
  out[a] = acc * (1.0f / (float)N_FREQ);
}

// ---------------------------------------------------------------------------
extern "C" void kernel_launch(void* const* d_in, const int* in_sizes, int n_in,
                              void* d_out, int out_size, void* d_ws, size_t ws_size,
                              hipStream_t stream) {
  const float* x = (const float*)d_in[0];   // (9600000, 4) float32
  float* out = (float*)d_out;               // (18001,) float32
  float* ws = (float*)d_ws;
  float* ws_X = ws;                         // 38*16 floats (X tiles, cos/sin cols)
  float* ws_h = ws + 1024;                  // 19*8 floats (tr + subdiag sums)

  music_dft_wmma<<<1, 128, 0, stream>>>(x, ws_X);
  music_eig<<<1, 32, 0, stream>>>(ws_X, ws_h);
  music_scan<<<(N_ANGLES + 255) / 256, 256, 0, stream>>>(ws_h, out);
}